// LabelSmoothingLoss_16621523435890
// MI455X (gfx1250) — compile-verified
//
#include <hip/hip_runtime.h>
#include <hip/hip_bf16.h>
#include <stdint.h>

// ---------------------------------------------------------------------------
// Label smoothing loss, MI455X (gfx1250).
//  loss = -sum_r [ 0.9*(out[r,tgt]-lse_r) + w_r * sum_{v in S'_r} (out[r,v]-lse_r) ]
//  lse_r = logsumexp over 32000 cols; S'_r = unique nonzero origin[seq, tpos:]
//  excluding target. Memory-bound: 262 MB / 23.3 TB/s ~ 11 us floor.
//  Data path: TDM tensor_load_to_lds double-buffered tiles (TENSORcnt),
//  consumed by 8 waves doing online (max, sum-exp). Final 256->1 reduction
//  uses accumulating V_WMMA_F32_16X16X4_F32.
// ---------------------------------------------------------------------------

#define CONF 0.9f
#define TILE 4096   // floats per TDM tile (16 KB); double buffer = 32 KB LDS

typedef __attribute__((ext_vector_type(4))) unsigned int v4u;
typedef __attribute__((ext_vector_type(8))) int          v8i;
typedef __attribute__((ext_vector_type(4))) int          v4i;
typedef __attribute__((ext_vector_type(2))) float        v2f;
typedef __attribute__((ext_vector_type(8))) float        v8f;

// Issue a 1-D TDM load of n_elems f32 from global address gaddr to LDS byte address lds_addr.
// D# per CDNA5 ISA: group0 = {count=1, lds_addr, global_addr, type=2},
// group1 = {data_size=4B, tensor_dim0=n, tensor_dim1=1, tile_dim0=n,
//           tile_dim1=1, stride0=n}. Groups 2/3 zero (unused dims).
__device__ __forceinline__ void tdm_load_1d(uint32_t lds_addr, uint64_t gaddr,
                                            int n_elems) {
  v4u g0;
  g0[0] = 1u;                                      // count=1, user descriptor
  g0[1] = lds_addr;                                // LDS byte address
  g0[2] = (uint32_t)gaddr;                         // global_addr[31:0]
  g0[3] = (uint32_t)(gaddr >> 32) | (2u << 30);    // global_addr[56:32] | type=2
  v8i g1;
  g1[0] = (int)(2u << 16);                         // workgroup_mask=0, data_size=2 (4B)
  g1[1] = (int)(((uint32_t)n_elems & 0xFFFFu) << 16);   // tensor_dim0[15:0]
  g1[2] = (int)(1u << 16);                         // tensor_dim0[31:16]=0, tensor_dim1=1
  g1[3] = (int)(((uint32_t)n_elems & 0xFFFFu) << 16);   // tile_dim0 = n
  g1[4] = 1;                                       // tile_dim1=1, tile_dim2=0
  g1[5] = n_elems;                                 // tensor_dim0_stride[31:0]
  g1[6] = 0;                                       // stride0 hi / stride1 lo
  g1[7] = 0;
  v4i z4; z4[0] = 0; z4[1] = 0; z4[2] = 0; z4[3] = 0;
  v8i z8;
  #pragma unroll
  for (int k = 0; k < 8; ++k) z8[k] = 0;
  // clang-23 / therock-10.0 lane: 6-arg form
  // (g0, g1, g2, g3, extra_group, cpol)
  __builtin_amdgcn_tensor_load_to_lds(g0, g1, z4, z4, z8, 0);
}

__global__ __launch_bounds__(256)
void row_loss_kernel(const float* __restrict__ out,
                     const int*  __restrict__ target,
                     const int*  __restrict__ target_len,
                     const int*  __restrict__ origin,
                     const int*  __restrict__ shard_p,
                     const int*  __restrict__ now_p,
                     float*      __restrict__ ws,
                     int V, int part, int max_len)
{
  __shared__ float buf[2][TILE];
  __shared__ int   sorigin[512];
  __shared__ float sm[8], ss[8];
  __shared__ float s_lse;

  const int tid  = threadIdx.x;
  const int lane = tid & 31;
  const int wid  = tid >> 5;
  const int r    = blockIdx.x;
  const uint64_t row_base =
      (uint64_t)(uintptr_t)out + (uint64_t)r * (uint64_t)V * 4ull;

  const int seq = r % part;
  const int ml  = (max_len < 512) ? max_len : 512;
  for (int j = tid; j < ml; j += 256)
    sorigin[j] = origin[seq * max_len + j];

  const int nt = (V + TILE - 1) / TILE;
  const bool issuer = (wid == 0);   // only wave 0 drives the TDM

  if (issuer) {
    int n0 = (V < TILE) ? V : TILE;
    tdm_load_1d((uint32_t)(uintptr_t)&buf[0][0], row_base, n0);
  }

  // Online (max, sum-exp) over the streamed row.
  float m = -3.4e38f, s = 0.f;
  for (int t = 0; t < nt; ++t) {
    const int cur = t & 1;
    const int rem = V - t * TILE;
    const int n   = (rem < TILE) ? rem : TILE;
    if (issuer) {
      if (t + 1 < nt) {
        int rem1 = V - (t + 1) * TILE;
        int n1   = (rem1 < TILE) ? rem1 : TILE;
        tdm_load_1d((uint32_t)(uintptr_t)&buf[cur ^ 1][0],
                    row_base + (uint64_t)(t + 1) * TILE * 4ull, n1);
        __builtin_amdgcn_s_wait_tensorcnt(1);  // tile t complete (in-order)
      } else {
        __builtin_amdgcn_s_wait_tensorcnt(0);
      }
    }
    __syncthreads();
    for (int i = tid; i < n; i += 256) {
      float x = buf[cur][i];
      if (x <= m) { s += __expf(x - m); }
      else        { s = s * __expf(m - x) + 1.0f; m = x; }
    }
    __syncthreads();
  }

  // Wave-level combine of (m, s), then cross-wave.
  for (int o = 16; o > 0; o >>= 1) {
    float m2 = __shfl_xor(m, o, 32);
    float s2 = __shfl_xor(s, o, 32);
    float M  = fmaxf(m, m2);
    s = s * __expf(m - M) + s2 * __expf(m2 - M);
    m = M;
  }
  if (lane == 0) { sm[wid] = m; ss[wid] = s; }
  __syncthreads();
  if (tid == 0) {
    float M = sm[0], S = ss[0];
    for (int k = 1; k < 8; ++k) {
      float m2 = sm[k], s2 = ss[k];
      float MM = fmaxf(M, m2);
      S = S * __expf(M - MM) + s2 * __expf(m2 - MM);
      M = MM;
    }
    s_lse = M + logf(S);
  }
  __syncthreads();

  // Sparse part: wave 0 only. Dedup origin[seq, tpos:] (scatter-SET semantics:
  // each unique column counted once), exclude col 0 and target column.
  if (wid == 0) {
    const float lse  = s_lse;
    const int tgt    = target[r];
    const int shard  = shard_p[0];
    const int nowv   = now_p[0];
    const int tpos   = nowv * shard + r / part;
    const int tl     = target_len[seq];
    const bool pad   = (tgt < 0 || tgt >= V);   // PADDING_IDX = -100
    const bool active = tpos < tl - 2;
    float w = 0.f;
    if (active) w = CONF / fmaxf((float)(tl - tpos - 2), 1.0f);

    float acc = 0.f;
    if (!pad && active) {
      for (int j = tpos + lane; j < ml; j += 32) {
        int v = sorigin[j];
        if (v == 0 || v == tgt) continue;
        bool first = true;
        for (int j2 = tpos; j2 < j; ++j2)
          if (sorigin[j2] == v) { first = false; break; }
        if (first) acc += out[(uint64_t)r * V + v] - lse;
      }
    }
    for (int o = 16; o > 0; o >>= 1) acc += __shfl_xor(acc, o, 32);
    if (lane == 0) {
      float loss = 0.f;
      if (!pad) {
        loss = CONF * (out[(uint64_t)r * V + tgt] - lse);
        if (active) loss += w * acc;
      }
      ws[r] = -loss;
    }
  }
}

// Deterministic final reduction: n partials -> 256 thread sums -> chain of
// accumulating V_WMMA_F32_16X16X4_F32 (B = ones so D[m][n] accumulates row
// sums of A); column 0 across lanes 0/16 holds the grand total.
__global__ __launch_bounds__(256)
void final_reduce_kernel(const float* __restrict__ ws,
                         float* __restrict__ out, int n)
{
  __shared__ float p[256];
  float a = 0.f;
  for (int i = threadIdx.x; i < n; i += 256) a += ws[i];
  p[threadIdx.x] = a;
  __syncthreads();

  if (threadIdx.x < 32) {            // full wave 0, EXEC all ones (WMMA req.)
    const int L = threadIdx.x;
    v8f c;
    #pragma unroll
    for (int k = 0; k < 8; ++k) c[k] = 0.f;
    v2f ones; ones[0] = 1.f; ones[1] = 1.f;
    #pragma unroll
    for (int t = 0; t < 4; ++t) {
      // A 16x4 f32 layout: lane L holds (M = L&15, K = 2*(L>>4)+{0,1})
      const int base = t * 64 + (L & 15) * 4 + 2 * (L >> 4);
      v2f a2; a2[0] = p[base]; a2[1] = p[base + 1];
      c = __builtin_amdgcn_wmma_f32_16x16x4_f32(
          /*neg_a=*/false, a2, /*neg_b=*/false, ones,
          /*c_mod=*/(short)0, c, /*reuse_a=*/false, /*reuse_b=*/false);
    }
    float col = c[0] + c[1] + c[2] + c[3] + c[4] + c[5] + c[6] + c[7];
    col += __shfl_xor(col, 16, 32);  // lanes 0..15: M=0..7, lanes 16..31: M=8..15
    if (L == 0) out[0] = col;
  }
}

extern "C" void kernel_launch(void* const* d_in, const int* in_sizes, int n_in,
                              void* d_out, int out_size, void* d_ws, size_t ws_size,
                              hipStream_t stream)
{
  const float* output     = (const float*)d_in[0];
  const int*   target     = (const int*)d_in[1];
  const int*   target_len = (const int*)d_in[2];
  const int*   origin     = (const int*)d_in[3];
  const int*   shard_p    = (const int*)d_in[4];
  // d_in[5] = part (value also known from in_sizes[2])
  const int*   now_p      = (const int*)d_in[6];

  const int B       = in_sizes[1];
  const int V       = in_sizes[0] / B;
  const int part    = in_sizes[2];
  const int max_len = in_sizes[3] / part;
  float* ws = (float*)d_ws;

  row_loss_kernel<<<B, 256, 0, stream>>>(output, target, target_len, origin,
                                         shard_p, now_p, ws, V, part, max_len);
  final_reduce_kernel<<<1, 256, 0, stream>>>(ws, (float*)d_out, B);
}